// GCNLayer_45973329936465
// MI455X (gfx1250) — compile-verified
//
#include <hip/hip_runtime.h>
#include <hip/hip_bf16.h>

typedef __attribute__((ext_vector_type(2))) float v2f;
typedef __attribute__((ext_vector_type(8))) float v8f;

#define IN_FEAT 128
#define OUT_FEAT 128

// ---------------------------------------------------------------------------
// Kernel 1: zero the output accumulator, init degree with the self-loop (1.0)
// ---------------------------------------------------------------------------
__global__ void gcn_init_kernel(float* __restrict__ out, float* __restrict__ deg,
                                int n_nodes, int total) {
    int idx = blockIdx.x * blockDim.x + threadIdx.x;
    if (idx < total) out[idx] = 0.0f;
    if (idx < n_nodes) deg[idx] = 1.0f;   // self-loop counted at dst
}

// ---------------------------------------------------------------------------
// Kernel 2: degree accumulation at dst (hardware fp32 atomics)
// ---------------------------------------------------------------------------
__global__ void gcn_deg_kernel(const int* __restrict__ dst, float* __restrict__ deg, int E) {
    int e = blockIdx.x * blockDim.x + threadIdx.x;
    if (e < E) unsafeAtomicAdd(&deg[dst[e]], 1.0f);
}

// ---------------------------------------------------------------------------
// Kernel 3: dinv = rsqrt(deg) (deg >= 1 always due to self loops)
// ---------------------------------------------------------------------------
__global__ void gcn_dinv_kernel(const float* __restrict__ deg, float* __restrict__ dinv, int n) {
    int i = blockIdx.x * blockDim.x + threadIdx.x;
    if (i < n) {
        float d = deg[i];
        dinv[i] = (d > 0.0f) ? rsqrtf(d) : 0.0f;
    }
}

// ---------------------------------------------------------------------------
// Kernel 4: h = x @ W using V_WMMA_F32_16X16X4_F32.
// One wave -> one 16x16 tile of h. Block = 256 threads = 8 waves covering
// the full 128-wide output for one 16-row band; grid.x = n_nodes/16 bands.
//
// A (16x4 f32): lane l holds A[l%16][(l/16)*2 + v], v=0..1  -> float2 load
// B (4x16 f32): lane l holds B[(l/16)*2 + v][l%16],  v=0..1
// C (16x16 f32, 8 VGPRs): lane l, vgpr v -> C[v + (l/16)*8][l%16]
// ---------------------------------------------------------------------------
__global__ void gcn_gemm_wmma_kernel(const float* __restrict__ x,
                                     const float* __restrict__ W,
                                     float* __restrict__ h) {
    const int lane = threadIdx.x & 31;
    const int wave = threadIdx.x >> 5;
    const int row0 = blockIdx.x * 16;     // M band
    const int n0   = wave * 16;           // N tile (8 waves * 16 = 128)
    const int m    = lane & 15;           // A row / B,C col within tile
    const int kgrp = lane >> 4;           // 0 or 1

    v8f c = {};
    const float* arow = x + (size_t)(row0 + m) * IN_FEAT + kgrp * 2;
    const float* bcol = W + (size_t)(kgrp * 2) * OUT_FEAT + n0 + m;

    #pragma unroll 8
    for (int k = 0; k < IN_FEAT; k += 4) {
        v2f a = *(const v2f*)(arow + k);
        v2f b;
        b.x = bcol[(size_t)(k + 0) * OUT_FEAT];
        b.y = bcol[(size_t)(k + 1) * OUT_FEAT];
        c = __builtin_amdgcn_wmma_f32_16x16x4_f32(
                /*neg_a=*/false, a, /*neg_b=*/false, b,
                /*c_mod=*/(short)0, c, /*reuse_a=*/false, /*reuse_b=*/false);
    }

    float* hout = h + (size_t)(row0 + kgrp * 8) * OUT_FEAT + n0 + m;
    #pragma unroll
    for (int v = 0; v < 8; ++v) hout[(size_t)v * OUT_FEAT] = c[v];
}

// Scalar tail for M not divisible by 16 (not used when n_nodes = 50000).
__global__ void gcn_gemm_tail_kernel(const float* __restrict__ x,
                                     const float* __restrict__ W,
                                     float* __restrict__ h,
                                     int row_start, int n_nodes) {
    int idx = blockIdx.x * blockDim.x + threadIdx.x;
    int row = row_start + (idx >> 7);
    int col = idx & 127;
    if (row >= n_nodes) return;
    float acc = 0.0f;
    for (int k = 0; k < IN_FEAT; ++k)
        acc += x[(size_t)row * IN_FEAT + k] * W[(size_t)k * OUT_FEAT + col];
    h[(size_t)row * OUT_FEAT + col] = acc;
}

// ---------------------------------------------------------------------------
// Kernel 5: per-edge gather/scale/scatter-add. One wave per edge, each lane
// handles 4 contiguous floats (float4 gather + 4 hardware fp32 atomics).
// ---------------------------------------------------------------------------
__global__ void gcn_scatter_kernel(const float* __restrict__ h,
                                   const float* __restrict__ dinv,
                                   const int* __restrict__ src,
                                   const int* __restrict__ dst,
                                   float* __restrict__ out, int E) {
    int gid  = blockIdx.x * blockDim.x + threadIdx.x;
    int e    = gid >> 5;
    int lane = gid & 31;
    if (e >= E) return;
    int s = src[e];
    int d = dst[e];
    float nrm = dinv[s] * dinv[d];
    const float4* hp = (const float4*)(h + (size_t)s * OUT_FEAT);
    float4 v = hp[lane];
    float* op = out + (size_t)d * OUT_FEAT + lane * 4;
    unsafeAtomicAdd(op + 0, v.x * nrm);
    unsafeAtomicAdd(op + 1, v.y * nrm);
    unsafeAtomicAdd(op + 2, v.z * nrm);
    unsafeAtomicAdd(op + 3, v.w * nrm);
}

// ---------------------------------------------------------------------------
// Kernel 6: self-loop contribution + bias + exact GELU, in place on out.
// ---------------------------------------------------------------------------
__global__ void gcn_finalize_kernel(float* __restrict__ out,
                                    const float* __restrict__ h,
                                    const float* __restrict__ dinv,
                                    const float* __restrict__ b, int total) {
    int idx = blockIdx.x * blockDim.x + threadIdx.x;
    if (idx >= total) return;
    int i = idx >> 7;          // node
    int f = idx & 127;         // feature
    float di = dinv[i];
    float v = out[idx] + h[idx] * di * di + b[f];
    // exact GELU: 0.5*v*(1+erf(v/sqrt(2)))
    out[idx] = 0.5f * v * (1.0f + erff(v * 0.70710678118654752440f));
}

// ---------------------------------------------------------------------------
extern "C" void kernel_launch(void* const* d_in, const int* in_sizes, int n_in,
                              void* d_out, int out_size, void* d_ws, size_t ws_size,
                              hipStream_t stream) {
    const float* x  = (const float*)d_in[0];   // [N, 128]
    const int*   ei = (const int*)d_in[1];     // [2, E]
    const float* W  = (const float*)d_in[2];   // [128, 128]
    const float* b  = (const float*)d_in[3];   // [128]
    float* out = (float*)d_out;                // [N, 128]

    const int n_nodes = in_sizes[0] / IN_FEAT;
    const int E       = in_sizes[1] / 2;
    const int* src = ei;
    const int* dst = ei + E;

    // scratch layout: h [N*128] | deg [N] | dinv [N]
    float* h    = (float*)d_ws;
    float* deg  = h + (size_t)n_nodes * OUT_FEAT;
    float* dinv = deg + n_nodes;

    const int total = n_nodes * OUT_FEAT;

    gcn_init_kernel<<<(total + 255) / 256, 256, 0, stream>>>(out, deg, n_nodes, total);
    gcn_deg_kernel<<<(E + 255) / 256, 256, 0, stream>>>(dst, deg, E);
    gcn_dinv_kernel<<<(n_nodes + 255) / 256, 256, 0, stream>>>(deg, dinv, n_nodes);

    const int m_tiles = n_nodes / 16;
    if (m_tiles > 0)
        gcn_gemm_wmma_kernel<<<m_tiles, 256, 0, stream>>>(x, W, h);
    const int tail_rows = n_nodes - m_tiles * 16;
    if (tail_rows > 0) {
        int tail_elems = tail_rows * OUT_FEAT;
        gcn_gemm_tail_kernel<<<(tail_elems + 255) / 256, 256, 0, stream>>>(
            x, W, h, m_tiles * 16, n_nodes);
    }

    long scatter_threads = (long)E * 32;
    gcn_scatter_kernel<<<(int)((scatter_threads + 255) / 256), 256, 0, stream>>>(
        h, dinv, src, dst, out, E);

    gcn_finalize_kernel<<<(total + 255) / 256, 256, 0, stream>>>(out, h, dinv, b, total);
}